// LoRAGenerator_14328010900075
// MI455X (gfx1250) — compile-verified
//
#include <hip/hip_runtime.h>
#include <hip/hip_bf16.h>
#include <math.h>

// ---------------- problem constants (from the reference) ----------------
#define EMB      384
#define PED      512
#define NUM_L    32
#define NTYPE    7
#define RANK     16
#define PCOUNT   (NUM_L * NTYPE)          // 224
#define PE_COLS  (PCOUNT * PED)           // 114688
#define ROWS     (2 * PCOUNT)             // 448 decoder rows (B=2)
#define HID1     256
#define HID2     1024                     // RANK*64
#define LAYER_SZ 1150976                  // floats per (b, layer) output block

typedef __attribute__((ext_vector_type(2))) float v2f;
typedef __attribute__((ext_vector_type(4))) float v4f;   // native vector: OK for NT builtins
typedef __attribute__((ext_vector_type(8))) float v8f;

// ---------------------------------------------------------------------------
// Kernel 1: pe = condition @ W_proj + b_proj   (2 x 114688), K = 384
// HBM-bound on the 176MB W_proj stream -> NT b128 loads, 4 columns/thread.
// ---------------------------------------------------------------------------
__global__ __launch_bounds__(256)
void pe_gemm_kernel(const float* __restrict__ cond, const float* __restrict__ W,
                    const float* __restrict__ bias, float* __restrict__ pe) {
    __shared__ float c0[EMB];
    __shared__ float c1[EMB];
    for (int i = threadIdx.x; i < EMB; i += 256) {
        c0[i] = cond[i];
        c1[i] = cond[EMB + i];
    }
    __syncthreads();

    const int j0 = (blockIdx.x * 256 + threadIdx.x) * 4;
    v4f a0 = {0.f, 0.f, 0.f, 0.f};
    v4f a1 = {0.f, 0.f, 0.f, 0.f};
    const v4f* wp = (const v4f*)(W + j0);
    const int wstride = PE_COLS / 4;  // v4f row stride
    #pragma unroll 4
    for (int k = 0; k < EMB; ++k) {
        v4f w = __builtin_nontemporal_load(wp + (size_t)k * wstride);
        float s0 = c0[k], s1 = c1[k];
        a0.x = fmaf(s0, w.x, a0.x); a0.y = fmaf(s0, w.y, a0.y);
        a0.z = fmaf(s0, w.z, a0.z); a0.w = fmaf(s0, w.w, a0.w);
        a1.x = fmaf(s1, w.x, a1.x); a1.y = fmaf(s1, w.y, a1.y);
        a1.z = fmaf(s1, w.z, a1.z); a1.w = fmaf(s1, w.w, a1.w);
    }
    v4f bv = *(const v4f*)(bias + j0);
    a0 += bv;
    a1 += bv;
    *(v4f*)(pe + j0)           = a0;
    *(v4f*)(pe + PE_COLS + j0) = a1;
}

// ---------------------------------------------------------------------------
// Kernel 2: fp32 WMMA GEMM  C = act(A(MxK) @ B(KxN) + bias), one wave per
// 16x16 D tile, V_WMMA_F32_16X16X4_F32 over K in steps of 4.
//   A frag: lane<16 -> row, {k,k+1}; lane>=16 -> row, {k+2,k+3}  (float2)
//   B frag: lane<16 -> col, {k,k+1}; lane>=16 -> col, {k+2,k+3}
//   D:      vgpr v -> row v (+8 for upper half-wave), col = lane&15
// ---------------------------------------------------------------------------
template <int GELU>
__global__ __launch_bounds__(32)
void gemm_wmma_f32(const float* __restrict__ A, const float* __restrict__ B,
                   const float* __restrict__ bias, float* __restrict__ C,
                   int M, int K, int N) {
    const int lane = threadIdx.x & 31;
    const int half = lane >> 4;
    const int l16  = lane & 15;
    const int tm   = blockIdx.y;
    const int tn   = blockIdx.x;

    const int rowA = tm * 16 + l16;
    const int colB = tn * 16 + l16;

    const float* aptr = A + (size_t)rowA * K + 2 * half;
    const float* bptr = B + (size_t)(2 * half) * N + colB;

    v8f c = {};
    for (int k = 0; k < K; k += 4) {
        v2f a, b;
        a.x = aptr[k];
        a.y = aptr[k + 1];
        b.x = bptr[(size_t)k * N];
        b.y = bptr[(size_t)(k + 1) * N];
        c = __builtin_amdgcn_wmma_f32_16x16x4_f32(false, a, false, b,
                                                  (short)0, c, false, false);
    }

    const float bv = bias[colB];
    #pragma unroll
    for (int v = 0; v < 8; ++v) {
        float val = c[v] + bv;
        if (GELU) val = 0.5f * val * (1.0f + erff(val * 0.70710678118654752f));
        const int row = tm * 16 + half * 8 + v;
        C[(size_t)row * N + colB] = val;
    }
}

// ---------------------------------------------------------------------------
// Kernel 3: expansion / tiling / scaling. 295MB of NT b128 stores; sources
// are tiny L2-resident tiles. Piece boundaries are all multiples of 16, so an
// aligned float4 output group always maps to one aligned float4 source group.
// ---------------------------------------------------------------------------
__global__ __launch_bounds__(256)
void expand_kernel(const float* __restrict__ Ab, const float* __restrict__ Bb,
                   const float* __restrict__ scales, float* __restrict__ out) {
    // per-layer piece start offsets (floats): A/B interleaved for 7 proj types
    constexpr int kStart[14] = {
        0,       65536,  131072, 196608, 212992, 278528, 294912,
        360448,  425984, 491520, 667648, 733184, 909312, 1085440};

    const int bl    = blockIdx.y;      // b*32 + layer
    const int layer = bl & 31;
    const int e     = (blockIdx.x * 256 + threadIdx.x) * 4;
    if (e >= LAYER_SZ) return;

    int seg = 0, segstart = 0;
    #pragma unroll
    for (int i = 1; i < 14; ++i)
        if (e >= kStart[i]) { seg = i; segstart = kStart[i]; }

    const int t = seg >> 1;
    const int p = e - segstart;

    const float* src;
    float s;
    if (!(seg & 1)) {                       // A piece: (RANK x in_d), tile cols mod 64
        int r, col;
        if (t == 6) { r = p / 11008; col = p - r * 11008; }   // down_proj in=11008
        else        { r = p >> 12;   col = p & 4095; }        // in=4096
        src = Ab + (((size_t)(bl * NTYPE + t) * RANK + r) << 6) + (col & 63);
        s   = scales[(layer * NTYPE + t) * 2 + 0];
    } else {                                // B piece: (out_d x RANK), tile rows mod 64
        const int row = p >> 4;
        const int r2  = p & 15;
        src = Bb + (((size_t)(bl * NTYPE + t) * 64 + (row & 63)) << 4) + r2;
        s   = scales[(layer * NTYPE + t) * 2 + 1];
    }

    v4f v = *(const v4f*)src;
    v *= s;
    __builtin_nontemporal_store(v, (v4f*)(out + (size_t)bl * LAYER_SZ + e));
}

// ---------------------------------------------------------------------------
extern "C" void kernel_launch(void* const* d_in, const int* in_sizes, int n_in,
                              void* d_out, int out_size, void* d_ws, size_t ws_size,
                              hipStream_t stream) {
    const float* cond   = (const float*)d_in[0];
    const float* W_proj = (const float*)d_in[1];
    const float* b_proj = (const float*)d_in[2];
    const float* WA1    = (const float*)d_in[3];
    const float* bA1    = (const float*)d_in[4];
    const float* WA2    = (const float*)d_in[5];
    const float* bA2    = (const float*)d_in[6];
    const float* WB1    = (const float*)d_in[7];
    const float* bB1    = (const float*)d_in[8];
    const float* WB2    = (const float*)d_in[9];
    const float* bB2    = (const float*)d_in[10];
    const float* scales = (const float*)d_in[11];
    float* out = (float*)d_out;

    // workspace layout (floats): pe | hA | hB | A_base | B_base  (~5.5 MB)
    float* pe = (float*)d_ws;
    float* hA = pe + 2 * PE_COLS;                 // 448 x 256
    float* hB = hA + ROWS * HID1;                 // 448 x 256
    float* Ab = hB + ROWS * HID1;                 // 448 x 1024
    float* Bb = Ab + ROWS * HID2;                 // 448 x 1024

    // 1) pe = cond @ W_proj + b  (HBM-bound stream of W_proj)
    pe_gemm_kernel<<<dim3(PE_COLS / 1024), dim3(256), 0, stream>>>(cond, W_proj, b_proj, pe);

    // 2) decoders: fp32 WMMA GEMMs (Linear-GELU-Linear, A and B paths)
    gemm_wmma_f32<1><<<dim3(HID1 / 16, ROWS / 16), dim3(32), 0, stream>>>(pe, WA1, bA1, hA, ROWS, PED, HID1);
    gemm_wmma_f32<1><<<dim3(HID1 / 16, ROWS / 16), dim3(32), 0, stream>>>(pe, WB1, bB1, hB, ROWS, PED, HID1);
    gemm_wmma_f32<0><<<dim3(HID2 / 16, ROWS / 16), dim3(32), 0, stream>>>(hA, WA2, bA2, Ab, ROWS, HID1, HID2);
    gemm_wmma_f32<0><<<dim3(HID2 / 16, ROWS / 16), dim3(32), 0, stream>>>(hB, WB2, bB2, Bb, ROWS, HID1, HID2);

    // 3) tile/scale expansion: 295 MB of non-temporal stores
    expand_kernel<<<dim3(LAYER_SZ / 1024, 64), dim3(256), 0, stream>>>(Ab, Bb, scales, out);
}